// SAT_80711025426642
// MI455X (gfx1250) — compile-verified
//
#include <hip/hip_runtime.h>

typedef float v2f __attribute__((ext_vector_type(2)));
typedef float v8f __attribute__((ext_vector_type(8)));

#define NN 4096
#define EE 131072
#define INDIM 32
#define DD 64
#define HH 8
#define DHH 8
#define FF 64
#define LL 4
#define ND (NN * DD)

// ---------------- WMMA helper: D = A(16x4) * B(4x16) + C, fp32 ----------------
__device__ __forceinline__ v8f wmma4(v2f a, v2f b, v8f c) {
  return __builtin_amdgcn_wmma_f32_16x16x4_f32(
      /*neg_a=*/false, a, /*neg_b=*/false, b,
      /*c_mod=*/(short)0, c, /*reuse_a=*/false, /*reuse_b=*/false);
}

// ---------------- small utility kernels ----------------
__global__ void zero_f(float* p, int n) {
  int i = blockIdx.x * blockDim.x + threadIdx.x;
  if (i < n) p[i] = 0.f;
}
__global__ void zero_i(int* p, int n) {
  int i = blockIdx.x * blockDim.x + threadIdx.x;
  if (i < n) p[i] = 0;
}
__global__ void count_deg(const int* __restrict__ src, const int* __restrict__ dst,
                          int* cs, int* ci) {
  int e = blockIdx.x * blockDim.x + threadIdx.x;
  if (e < EE) {
    atomicAdd(&cs[src[e]], 1);
    atomicAdd(&ci[dst[e]], 1);
  }
}
__global__ void finalize_deg(const int* __restrict__ cs, const int* __restrict__ ci,
                             float* degree, float* invdegin) {
  int n = blockIdx.x * blockDim.x + threadIdx.x;
  if (n < NN) {
    degree[n] = rsqrtf(1.f + (float)cs[n]);
    invdegin[n] = 1.f / fmaxf((float)ci[n], 1.f);
  }
}
// msg[dst] += h[src]  (per edge, per channel)
__global__ void scatter_msg(const float* __restrict__ h, const int* __restrict__ src,
                            const int* __restrict__ dst, float* msg) {
  int i = blockIdx.x * blockDim.x + threadIdx.x;
  if (i < EE * DD) {
    int e = i >> 6, c = i & 63;
    atomicAdd(&msg[dst[e] * DD + c], h[src[e] * DD + c]);
  }
}
// out = h + msg * invdegin[row]
__global__ void preadd(const float* __restrict__ h, const float* __restrict__ msg,
                       const float* __restrict__ invdegin, float* out) {
  int i = blockIdx.x * blockDim.x + threadIdx.x;
  if (i < ND) out[i] = h[i] + msg[i] * invdegin[i >> 6];
}

// ---------------- generic WMMA GEMM: C[nrows x 64] = act(A[nrows x kdim] @ B[kdim x 64])
// grid.x = nrows/16, block = 128 (4 waves -> 4 column tiles of 16)
__global__ void __launch_bounds__(128) gemm_k(const float* __restrict__ A,
                                              const float* __restrict__ B,
                                              float* __restrict__ C,
                                              int kdim, int relu) {
  int wave = threadIdx.x >> 5;       // column tile 0..3
  int lane = threadIdx.x & 31;
  int rowbase = blockIdx.x * 16;
  int colbase = wave * 16;
  int m = lane & 15;
  int khalf = (lane >> 4) * 2;       // 0 for lanes 0-15, 2 for lanes 16-31
  v8f c = {};
  for (int k0 = 0; k0 < kdim; k0 += 4) {
    v2f a, b;
    const float* ar = A + (rowbase + m) * kdim + k0 + khalf;
    a.x = ar[0];
    a.y = ar[1];
    b.x = B[(k0 + khalf) * 64 + colbase + m];
    b.y = B[(k0 + khalf + 1) * 64 + colbase + m];
    c = wmma4(a, b, c);
  }
  // C layout: VGPR j -> row = j + 8*(lane>>4), col = lane&15
  int rr = rowbase + 8 * (lane >> 4);
  int cc = colbase + m;
#pragma unroll
  for (int j = 0; j < 8; j++) {
    float v = c[j];
    if (relu) v = fmaxf(v, 0.f);
    C[(rr + j) * 64 + cc] = v;
  }
}

// ---------------- flash attention: block = 256 (wave == head), grid = N/16 query tiles
__global__ void __launch_bounds__(256) attn_kernel(const float* __restrict__ q,
                                                   const float* __restrict__ k,
                                                   const float* __restrict__ v,
                                                   float* __restrict__ o) {
  __shared__ float pt[HH][16 * 17];  // per-wave P-tile scratch (padded rows)
  int h = threadIdx.x >> 5;          // head = wave id
  int lane = threadIdx.x & 31;
  int qbase = blockIdx.x * 16;
  int m = lane & 15;
  int half = lane >> 4;
  int khalf = half * 2;
  const float scale = 0.35355339059327373f;  // 1/sqrt(DH)

  // preload q tile as A operands (K = 0..3 and 4..7), scale folded in
  const float* qrow = q + (qbase + m) * DD + h * DHH;
  v2f qa0, qa1;
  qa0.x = qrow[khalf] * scale;
  qa0.y = qrow[khalf + 1] * scale;
  qa1.x = qrow[4 + khalf] * scale;
  qa1.y = qrow[4 + khalf + 1] * scale;

  float mrow[8], lrow[8];
#pragma unroll
  for (int j = 0; j < 8; j++) { mrow[j] = -1e30f; lrow[j] = 0.f; }
  v8f acc = {};
  float* myp = &pt[h][0];

  for (int kb = 0; kb < NN; kb += 16) {
    // ---- scores S(16x16) = q_tile @ k_tile^T : B[d][key], key = m
    v8f s = {};
    const float* krow = k + (kb + m) * DD + h * DHH;
    v2f b;
    b.x = krow[khalf];     b.y = krow[khalf + 1];
    s = wmma4(qa0, b, s);
    b.x = krow[4 + khalf]; b.y = krow[4 + khalf + 1];
    s = wmma4(qa1, b, s);

    // ---- online softmax; each VGPR j holds row (j + 8*half) across 16 lanes
#pragma unroll
    for (int j = 0; j < 8; j++) {
      float sm = s[j];
#pragma unroll
      for (int off = 1; off < 16; off <<= 1) sm = fmaxf(sm, __shfl_xor(sm, off, 32));
      float mn = fmaxf(mrow[j], sm);
      float alpha = __expf(mrow[j] - mn);
      mrow[j] = mn;
      float p = __expf(s[j] - mn);
      float ps = p;
#pragma unroll
      for (int off = 1; off < 16; off <<= 1) ps += __shfl_xor(ps, off, 32);
      lrow[j] = lrow[j] * alpha + ps;
      acc[j] *= alpha;
      myp[(j + 8 * half) * 17 + m] = p;  // stash P in C-layout -> plain row-major
    }
    asm volatile("s_wait_dscnt 0" ::: "memory");  // same-wave DS ordering + compiler barrier

    // ---- acc += P(16x16) @ V_tile(16keys x 8d)  (B cols 8..15 are zero padding)
#pragma unroll
    for (int kc = 0; kc < 16; kc += 4) {
      v2f a, bb;
      a.x = myp[m * 17 + kc + khalf];
      a.y = myp[m * 17 + kc + khalf + 1];
      int key0 = kb + kc + khalf;
      bb.x = (m < DHH) ? v[key0 * DD + h * DHH + m] : 0.f;
      bb.y = (m < DHH) ? v[(key0 + 1) * DD + h * DHH + m] : 0.f;
      acc = wmma4(a, bb, acc);
    }
    asm volatile("s_wait_dscnt 0" ::: "memory");
  }

  // ---- write o[row, h*8 + d] = acc / l   (C layout: VGPR j -> row j+8*half, col m)
  if (m < DHH) {
#pragma unroll
    for (int j = 0; j < 8; j++)
      o[(qbase + j + 8 * half) * DD + h * DHH + m] = acc[j] / lrow[j];
  }
}

// ---------------- fused residual (+ optional row scale) + LayerNorm; 1 row / 64-thread block
__global__ void __launch_bounds__(64) resid_ln(float* __restrict__ h,
                                               const float* __restrict__ add,
                                               const float* __restrict__ rowscale,
                                               int use_scale,
                                               const float* __restrict__ g,
                                               const float* __restrict__ b) {
  __shared__ float red[64];
  int row = blockIdx.x, c = threadIdx.x;
  float sc = use_scale ? rowscale[row] : 1.f;
  float val = h[row * DD + c] + add[row * DD + c] * sc;
  red[c] = val;
  __syncthreads();
  for (int s = 32; s > 0; s >>= 1) { if (c < s) red[c] += red[c + s]; __syncthreads(); }
  float mean = red[0] * (1.f / 64.f);
  __syncthreads();
  float d = val - mean;
  red[c] = d * d;
  __syncthreads();
  for (int s = 32; s > 0; s >>= 1) { if (c < s) red[c] += red[c + s]; __syncthreads(); }
  float var = red[0] * (1.f / 64.f);
  h[row * DD + c] = d * rsqrtf(var + 1e-5f) * g[c] + b[c];
}

// ---------------- host orchestration ----------------
extern "C" void kernel_launch(void* const* d_in, const int* in_sizes, int n_in,
                              void* d_out, int out_size, void* d_ws, size_t ws_size,
                              hipStream_t stream) {
  const float* x  = (const float*)d_in[0];
  const int*  ei  = (const int*)d_in[1];
  const int*  src = ei;
  const int*  dst = ei + EE;
  const float* We = (const float*)d_in[2];
  const float* Wg = (const float*)d_in[3];
  const float* Wq = (const float*)d_in[4];
  const float* Wk = (const float*)d_in[5];
  const float* Wv = (const float*)d_in[6];
  const float* Wo = (const float*)d_in[7];
  const float* W1 = (const float*)d_in[8];
  const float* W2 = (const float*)d_in[9];
  const float* g1 = (const float*)d_in[10];
  const float* b1 = (const float*)d_in[11];
  const float* g2 = (const float*)d_in[12];
  const float* b2 = (const float*)d_in[13];

  float* h = (float*)d_out;  // h lives in d_out; fully written by embed GEMM
  float* ws = (float*)d_ws;
  float* msg   = ws;
  float* sbuf  = msg + ND;
  float* sfeat = sbuf + ND;
  float* qb    = sfeat + ND;
  float* kb    = qb + ND;
  float* vb    = kb + ND;
  float* oatt  = vb + ND;
  float* t1    = oatt + ND;
  float* t2    = t1 + ND;
  float* degree   = t2 + ND;
  float* invdegin = degree + NN;
  int*   cnt      = (int*)(invdegin + NN);  // 2*NN ints

  // degrees
  zero_i<<<(2 * NN + 255) / 256, 256, 0, stream>>>(cnt, 2 * NN);
  count_deg<<<(EE + 255) / 256, 256, 0, stream>>>(src, dst, cnt, cnt + NN);
  finalize_deg<<<(NN + 255) / 256, 256, 0, stream>>>(cnt, cnt + NN, degree, invdegin);

  // h = x @ We
  gemm_k<<<NN / 16, 128, 0, stream>>>(x, We, h, INDIM, 0);

  for (int l = 0; l < LL; l++) {
    // GNN structure extractor
    zero_f<<<(ND + 255) / 256, 256, 0, stream>>>(msg, ND);
    scatter_msg<<<(EE * DD + 255) / 256, 256, 0, stream>>>(h, src, dst, msg);
    preadd<<<(ND + 255) / 256, 256, 0, stream>>>(h, msg, invdegin, sbuf);
    gemm_k<<<NN / 16, 128, 0, stream>>>(sbuf, Wg + l * DD * DD, sfeat, DD, 1);

    // projections
    gemm_k<<<NN / 16, 128, 0, stream>>>(sfeat, Wq + l * DD * DD, qb, DD, 0);
    gemm_k<<<NN / 16, 128, 0, stream>>>(sfeat, Wk + l * DD * DD, kb, DD, 0);
    gemm_k<<<NN / 16, 128, 0, stream>>>(h,     Wv + l * DD * DD, vb, DD, 0);

    // global attention (flash streaming)
    attn_kernel<<<NN / 16, 256, 0, stream>>>(qb, kb, vb, oatt);

    // output proj * degree, residual + LN
    gemm_k<<<NN / 16, 128, 0, stream>>>(oatt, Wo + l * DD * DD, t1, DD, 0);
    resid_ln<<<NN, 64, 0, stream>>>(h, t1, degree, 1, g1 + l * DD, b1 + l * DD);

    // FFN + residual + LN
    gemm_k<<<NN / 16, 128, 0, stream>>>(h, W1 + l * DD * FF, t1, DD, 1);
    gemm_k<<<NN / 16, 128, 0, stream>>>(t1, W2 + l * FF * DD, t2, FF, 0);
    resid_ln<<<NN, 64, 0, stream>>>(h, t2, nullptr, 0, g2 + l * DD, b2 + l * DD);
  }
}